// SparseLinear_85444079387040
// MI455X (gfx1250) — compile-verified
//
#include <hip/hip_runtime.h>
#include <stdint.h>

typedef float v2f __attribute__((ext_vector_type(2)));
typedef float v8f __attribute__((ext_vector_type(8)));

#define M_DIM 4096
#define K_DIM 4096
#define N_DIM 8192

#define BM 128
#define BN 128
#define BK 16
#define LDA  20    // As row stride (floats): rows 16B aligned, frag reads conflict-free
#define LDBP 288   // Bs pair-row stride (floats): 288 % 64 == 32 -> K-halves on disjoint banks
#define NBUF 3     // depth-2 async pipeline

#define ABUF_BYTES (BM * LDA * 4)          // 10240
#define BBUF_BYTES ((BK / 2) * LDBP * 4)   // 9216

__global__ __launch_bounds__(256, 1)
void sparse_linear_wmma_f32(const float* __restrict__ Wm,
                            const float* __restrict__ Xm,
                            float* __restrict__ Ym) {
    // As: [m][k], row stride LDA.  Bs: (k,n) -> (k>>1)*LDBP + n*2 + (k&1)
    __shared__ __align__(16) float As[NBUF][BM * LDA];
    __shared__ __align__(16) float Bs[NBUF][(BK / 2) * LDBP];

    const int tid    = threadIdx.x;
    const int lane   = tid & 31;
    const int wave   = tid >> 5;
    const int wr     = wave >> 1;            // 0..3: 32 M-rows each
    const int wc     = wave & 1;             // 0..1: 64 N-cols each
    const int laneLo = lane & 15;
    const int hiSel  = lane >> 4;            // 0 | 1
    const int kHalf  = hiSel << 1;           // A frag K base: 0 | 2
    const int rBase  = hiSel << 3;           // C/D row offset: 0 | 8

    const int blockN = blockIdx.x * BN;
    const int blockM = blockIdx.y * BM;

    // ---- async-copy descriptors (4 VGPRs; segment deltas are compile-time) ----
    // A: thread covers 16B segs: seg i: row = tid>>2 + i*64, col4 = (tid&3)*4
    const int rA0 = tid >> 2;
    const int cA0 = (tid & 3) << 2;
    const uint32_t aVb = (uint32_t)((rA0 * K_DIM + cA0) * 4);                 // global byte off
    const uint32_t aLb = (uint32_t)(uintptr_t)&As[0][rA0 * LDA + cA0];        // LDS byte addr
    // B: thread covers 4B elems: seg j: k = 2j + (tid>>7), n = tid&127
    const int t7  = tid >> 7;
    const int nB0 = tid & 127;
    const uint32_t bVb = (uint32_t)((t7 * N_DIM + nB0) * 4);
    const uint32_t bLb = (uint32_t)(uintptr_t)&Bs[0][nB0 * 2 + t7];

    const uint64_t aS0 = (uint64_t)(uintptr_t)Wm + (uint64_t)blockM * K_DIM * 4;
    const uint64_t bS0 = (uint64_t)(uintptr_t)Xm + (uint64_t)blockN * 4;

    auto issue_async = [&](int kc, int buf) {
        const uint64_t sA   = aS0 + (uint64_t)kc * (BK * 4);
        const uint64_t sB   = bS0 + (uint64_t)kc * (BK * N_DIM * 4);
        const uint32_t offA = (uint32_t)buf * ABUF_BYTES;
        const uint32_t offB = (uint32_t)buf * BBUF_BYTES;
#pragma unroll
        for (int i = 0; i < 2; ++i) {
            const uint32_t l = aLb + offA + (uint32_t)(i * 64 * LDA * 4);
            const uint32_t v = aVb + (uint32_t)(i * 64 * K_DIM * 4);
            asm volatile("global_load_async_to_lds_b128 %0, %1, %2"
                         :: "v"(l), "v"(v), "s"(sA) : "memory");
        }
#pragma unroll
        for (int j = 0; j < 8; ++j) {
            const uint32_t l = bLb + offB + (uint32_t)(j * LDBP * 4);
            const uint32_t v = bVb + (uint32_t)(j * 2 * N_DIM * 4);
            asm volatile("global_load_async_to_lds_b32 %0, %1, %2"
                         :: "v"(l), "v"(v), "s"(sB) : "memory");
        }
    };

    // ---- accumulators: 2 (M) x 4 (N) tiles of 16x16 f32 ----
    v8f acc[2][4];
#pragma unroll
    for (int mt = 0; mt < 2; ++mt)
#pragma unroll
        for (int nt = 0; nt < 4; ++nt)
            acc[mt][nt] = (v8f){0.f, 0.f, 0.f, 0.f, 0.f, 0.f, 0.f, 0.f};

    // ---- prologue: two chunks in flight ----
    issue_async(0, 0);
    issue_async(1, 1);

    const int NKC = K_DIM / BK;   // 256 chunks
    int bufc = 0;
    for (int kc = 0; kc < NKC; ++kc) {
        // <=10 outstanding == newest chunk only -> chunk kc has landed (in-order completion)
        asm volatile("s_wait_asynccnt 0xa" ::: "memory");
        __syncthreads();   // all waves' chunk-kc data visible; all waves done reading buf (kc-1)%3

        if (kc + 2 < NKC) {
            int bufn = bufc + 2; if (bufn >= NBUF) bufn -= NBUF;
            issue_async(kc + 2, bufn);
        }

        const float* A0 = As[bufc];
        const float* B0 = Bs[bufc];

#pragma unroll
        for (int ks = 0; ks < 4; ++ks) {
            const int kk = ks * 4 + kHalf;
            v2f af[2];
#pragma unroll
            for (int mt = 0; mt < 2; ++mt) {
                const int row = wr * 32 + mt * 16 + laneLo;
                af[mt] = *(const v2f*)&A0[row * LDA + kk];                     // ds_load_b64
            }
            v2f bf[4];
#pragma unroll
            for (int nt = 0; nt < 4; ++nt) {
                const int col = wc * 64 + nt * 16 + laneLo;
                bf[nt] = *(const v2f*)&B0[(ks * 2 + hiSel) * LDBP + col * 2];  // ds_load_b64
            }
#pragma unroll
            for (int mt = 0; mt < 2; ++mt)
#pragma unroll
                for (int nt = 0; nt < 4; ++nt)
                    acc[mt][nt] = __builtin_amdgcn_wmma_f32_16x16x4_f32(
                        false, af[mt], false, bf[nt],
                        (short)0, acc[mt][nt], false, false);
        }

        bufc = (bufc + 1 == NBUF) ? 0 : bufc + 1;
    }

    // ---- epilogue ----
#pragma unroll
    for (int mt = 0; mt < 2; ++mt) {
#pragma unroll
        for (int nt = 0; nt < 4; ++nt) {
            const int col = blockN + wc * 64 + nt * 16 + laneLo;
#pragma unroll
            for (int r = 0; r < 8; ++r) {
                const int row = blockM + wr * 32 + mt * 16 + rBase + r;
                Ym[(size_t)row * N_DIM + col] = acc[mt][nt][r];
            }
        }
    }
}

extern "C" void kernel_launch(void* const* d_in, const int* in_sizes, int n_in,
                              void* d_out, int out_size, void* d_ws, size_t ws_size,
                              hipStream_t stream) {
    const float* Wm = (const float*)d_in[0];   // [4096, 4096] row-major
    const float* Xm = (const float*)d_in[1];   // [4096, 8192] row-major
    float*       Ym = (float*)d_out;           // [4096, 8192] row-major

    dim3 grid(N_DIM / BN, M_DIM / BM);         // 64 x 32
    sparse_linear_wmma_f32<<<grid, 256, 0, stream>>>(Wm, Xm, Ym);
}